// MultinomialCELoss_67791763800580
// MI455X (gfx1250) — compile-verified
//
#include <hip/hip_runtime.h>
#include <hip/hip_bf16.h>

typedef __attribute__((ext_vector_type(2))) float v2f;
typedef __attribute__((ext_vector_type(8))) float v8f;

#define N_   8
#define Q_   441
#define H_   128
#define W_   128
#define NBINS_ 21

// out[w] = -sum_{n,h} log( x[n, bin(y[n,:,h,w]), h, w] )
//
// Reduction done as ones(16x4) x G(4x16) WMMA accumulation:
//   D = A(ones) * B(gathered log-x rows) + C  sums 4 (n,h) rows per step
// into 16 per-column (w) f32 accumulators. A = ones makes the result
// invariant to the exact K<->(vgpr,lane-half) layout of B, as long as each
// step feeds 4 distinct rows per column (lanes 0-15 get rows {r,r+1},
// lanes 16-31 get rows {r+2,r+3}).
__global__ __launch_bounds__(1024) void
MultinomialCELoss_67791763800580_kernel(const float* __restrict__ x,
                                        const float* __restrict__ y,
                                        float* __restrict__ out) {
    const int lane = threadIdx.x & 31;       // wave32
    const int wv   = threadIdx.x >> 5;       // 0..31 waves in block
    const int col  = lane & 15;              // w within tile
    const int half = lane >> 4;              // 0 or 1
    const int w    = blockIdx.x * 16 + col;  // global w column

    v8f acc = {};                            // f32 column accumulators
    v2f a;  a[0] = 1.0f; a[1] = 1.0f;        // A = ones (layout-invariant)

    const int rowStart = wv * 32;            // this wave's 32 (n,h) rows
#pragma unroll
    for (int i = 0; i < 8; ++i) {
        const int rbase = rowStart + i * 4 + 2 * half;
        v2f b;
#pragma unroll
        for (int j = 0; j < 2; ++j) {
            const int p = rbase + j;         // flat (n,h) row, 0..1023
            const int n = p >> 7;            // H_ = 128
            const int h = p & 127;

            // y: [N, 2, H, W]
            const float ya = y[(((n * 2 + 0) * H_) + h) * W_ + w];
            const float yb = y[(((n * 2 + 1) * H_) + h) * W_ + w];

            int q0 = (int)floorf((ya + 110.0f) / 10.0f);
            q0 = min(max(q0, 0), NBINS_ - 1);
            int q1 = (int)floorf((yb + 110.0f) / 10.0f);
            q1 = min(max(q1, 0), NBINS_ - 1);
            const int idx = q0 * NBINS_ + q1;

            // x: [N, Q, H, W] gather
            const float xv = x[(((n * Q_ + idx) * H_) + h) * W_ + w];
            b[j] = logf(xv);
        }
        // D = ones(16x4) * B(4x16) + C : folds 4 rows into column sums.
        acc = __builtin_amdgcn_wmma_f32_16x16x4_f32(
            /*neg_a=*/false, a, /*neg_b=*/false, b,
            /*c_mod=*/(short)0, acc, /*reuse_a=*/false, /*reuse_b=*/false);
    }

    // acc VGPR0, lanes 0..15 hold this wave's partial column sums
    // (all D rows are identical since A rows are identical).
    __shared__ float smem[32][16];
    if (lane < 16) smem[wv][lane] = acc[0];
    __syncthreads();

    // Deterministic fixed-order cross-wave combine; one writer per w.
    if (threadIdx.x < 16) {
        float t = 0.0f;
#pragma unroll
        for (int k = 0; k < 32; ++k) t += smem[k][threadIdx.x];
        out[blockIdx.x * 16 + threadIdx.x] = -t;
    }
}

extern "C" void kernel_launch(void* const* d_in, const int* in_sizes, int n_in,
                              void* d_out, int out_size, void* d_ws, size_t ws_size,
                              hipStream_t stream) {
    const float* x = (const float*)d_in[0];   // [8, 441, 128, 128] f32
    const float* y = (const float*)d_in[1];   // [8, 2, 128, 128]   f32
    float* out = (float*)d_out;               // [128] f32

    // 8 blocks: one 16-wide w tile each; 1024 threads = 32 waves per block.
    MultinomialCELoss_67791763800580_kernel<<<dim3(W_ / 16), dim3(1024), 0, stream>>>(x, y, out);
}